// TopKTopPSampler_19069654794869
// MI455X (gfx1250) — compile-verified
//
#include <hip/hip_runtime.h>
#include <hip/hip_bf16.h>
#include <math.h>

typedef unsigned int u32;
typedef unsigned long long u64;
typedef float vfloat4 __attribute__((ext_vector_type(4)));

#define BDIM    1024
#define NB1     2048    // level-1/2 buckets (11 bits); key split 11+11+10
#define NB3     1024    // level-3 buckets (10 bits)
#define CHUNK   16      // buckets per thread in the scan
#define AUXN    128     // NB1 / CHUNK
#define BUFN    9216    // shared candidate pool (floats, 36 KB)
#define TOPCAP  1792    // top-region capacity (grows down from BUFN)
#define CANDCAP 7424    // bucket-b1 region capacity (grows up from 0)

static_assert(TOPCAP + CANDCAP == BUFN, "buffer split");

// monotonic float -> u32 key (larger float => larger key)
__device__ __forceinline__ u32 fkey(float x) {
    u32 b = __float_as_uint(x);
    return (b & 0x80000000u) ? ~b : (b | 0x80000000u);
}

// In-place inclusive suffix sums over h[0..nb).
template <typename T>
__device__ __forceinline__ void suffixScanT(T* h, int nb, int tid, u64* a0, u64* a1) {
    const int nchunk = nb / CHUNK;
    if (tid < nchunk) {
        const int base = tid * CHUNK;
        u64 run = 0;
        #pragma unroll
        for (int j = CHUNK - 1; j >= 0; --j) { run += (u64)h[base + j]; h[base + j] = (T)run; }
        a0[tid] = run;
    } else if (tid < AUXN) {
        a0[tid] = 0ULL;
    }
    __syncthreads();
    u64* src = a0; u64* dst = a1;
    for (int off = 1; off < AUXN; off <<= 1) {
        if (tid < AUXN) {
            u64 v = src[tid];
            if (tid + off < AUXN) v += src[tid + off];
            dst[tid] = v;
        }
        __syncthreads();
        u64* t = src; src = dst; dst = t;
    }
    if (tid < nchunk) {
        const u64 addv = (tid + 1 < AUXN) ? src[tid + 1] : 0ULL;
        const int base = tid * CHUNK;
        #pragma unroll
        for (int j = 0; j < CHUNK; ++j) h[base + j] = (T)((u64)h[base + j] + addv);
    }
    __syncthreads();
}

// Find unique b with SS[b] >= T > SS[b+1] (SS = inclusive suffix sums in h).
template <typename T>
__device__ __forceinline__ void findBucketT(T* h, int nb, u64 Tt, int tid,
                                            u32* outB, u64* outNext) {
    if (tid == 0) { *outB = 0u; *outNext = 0ULL; }
    __syncthreads();
    const int nchunk = nb / CHUNK;
    if (tid < nchunk) {
        const int base = tid * CHUNK;
        #pragma unroll
        for (int j = 0; j < CHUNK; ++j) {
            const int b = base + j;
            const u64 ss = (u64)h[b];
            const u64 nx = (b + 1 < nb) ? (u64)h[b + 1] : 0ULL;
            if (ss >= Tt && nx < Tt) { *outB = (u32)b; *outNext = nx; }
        }
    }
    __syncthreads();
}

struct ShmT {
    u64 A[NB1];         // weighted histograms (L1, then reused for L2/L3)   (16 KB)
    u32 Cc[NB1];        // count histograms (L1/L2/L3)                        (8 KB)
    u64 auxA[AUXN];     // scan ping                                          (1 KB)
    u64 auxB[AUXN];     // scan pong                                          (1 KB)
    float buf[BUFN];    // [0,CANDCAP): bucket-b1 cands; [BUFN-TOPCAP,BUFN): top cands (36 KB)
    u32 candCnt, topCnt;
    u32 foundB;
    u64 foundNext;
    u32 maxKey;
    float M;
    u64 target;
};

__global__ __launch_bounds__(BDIM, 1)
void topktopp_mask_kernel(const float* __restrict__ logits,
                          const int*   __restrict__ krow,
                          const float* __restrict__ prow,
                          float*       __restrict__ outp,
                          int V)
{
    __shared__ ShmT s;
    const int row = blockIdx.x;
    const int tid = threadIdx.x;
    const float* __restrict__ rp = logits + (size_t)row * (size_t)V;
    float* __restrict__ op = outp + (size_t)row * (size_t)V;
    const bool vec4 = ((V & 3) == 0);
    const int n4 = V >> 2;
    const float SCALE = 1.099511627776e12f; // 2^40 fixed point for exp weights

    auto zeroA = [&](int nb) { for (int i = tid; i < nb; i += BDIM) s.A[i] = 0ULL; };
    auto zeroC = [&](int nb) { for (int i = tid; i < nb; i += BDIM) s.Cc[i] = 0u; };

    // Stream the row through a per-element functor (b128 loads, prefetch ahead).
    auto forEach = [&](auto f) {
        if (vec4) {
            const vfloat4* __restrict__ r4 = (const vfloat4*)rp;
            for (int i = tid; i < n4; i += BDIM) {
                if (i + BDIM < n4) __builtin_prefetch((const void*)(r4 + i + BDIM), 0, 3);
                vfloat4 v = r4[i];
                f(v.x); f(v.y); f(v.z); f(v.w);
            }
        } else {
            for (int i = tid; i < V; i += BDIM) f(rp[i]);
        }
    };

    // ---------- P1: row max + count histogram on key[31:21] (global read #1) ----------
    zeroC(NB1);
    if (tid == 0) s.maxKey = 0u;
    __syncthreads();
    {
        u32 lmax = 0u;
        forEach([&](float x) {
            const u32 u = fkey(x);
            lmax = lmax > u ? lmax : u;
            atomicAdd(&s.Cc[u >> 21], 1u);
        });
        atomicMax(&s.maxKey, lmax);
    }
    __syncthreads();
    if (tid == 0) {
        const u32 mk = s.maxKey;
        const u32 mb = (mk & 0x80000000u) ? (mk & 0x7FFFFFFFu) : ~mk;
        s.M = __uint_as_float(mb);
    }
    u64 kT = (u64)(krow[row] < 1 ? 1 : krow[row]);
    if (kT > (u64)V) kT = (u64)V;
    suffixScanT(s.Cc, NB1, tid, s.auxA, s.auxB);
    findBucketT(s.Cc, NB1, kT, tid, &s.foundB, &s.foundNext);
    const u32 b1  = s.foundB;
    const u64 kT2 = kT - s.foundNext;
    __syncthreads();
    // top-region cutoff: smallest bucket with suffix-count <= TOPCAP (exact, from counts)
    findBucketT(s.Cc, NB1, (u64)(TOPCAP + 1), tid, &s.foundB, &s.foundNext);
    const u32 bTop = s.foundB + 1u;
    __syncthreads();

    // ---------- P2: count L2 (bucket b1) + weighted L1 partial (buckets > b1)
    //              + compact bucket-b1 elems + compact top region (global read #2) ----------
    zeroC(NB1);
    zeroA(NB1);
    if (tid == 0) { s.candCnt = 0u; s.topCnt = 0u; }
    __syncthreads();
    const float M = s.M;
    forEach([&](float x) {
        const u32 u = fkey(x);
        const u32 b = u >> 21;
        if (b > b1) {
            // definitely kept regardless of the exact top-k threshold
            atomicAdd(&s.A[b], (u64)(__expf(x - M) * SCALE));
        } else if (b == b1) {
            atomicAdd(&s.Cc[(u >> 10) & (NB1 - 1)], 1u);
            const u32 idx = atomicAdd(&s.candCnt, 1u);
            if (idx < CANDCAP) s.buf[idx] = x;
        }
        if (b >= bTop) {
            const u32 it = atomicAdd(&s.topCnt, 1u);
            if (it < TOPCAP) s.buf[(BUFN - 1) - it] = x;
        }
    });
    __syncthreads();
    const int candN1 = (int)s.candCnt;   // true counts (may exceed caps)
    const int topN   = (int)s.topCnt;
    suffixScanT(s.Cc, NB1, tid, s.auxA, s.auxB);
    findBucketT(s.Cc, NB1, kT2, tid, &s.foundB, &s.foundNext);
    const u32 b2  = s.foundB;
    const u64 kT3 = kT2 - s.foundNext;
    const u32 pre22 = (b1 << 11) | b2;
    __syncthreads();

    // ---------- P3: count level 3 on key[9:0] -> exact T_k (LDS, global fallback) ----------
    zeroC(NB3);
    __syncthreads();
    if (candN1 <= CANDCAP) {
        for (int i = tid; i < candN1; i += BDIM) {
            const u32 u = fkey(s.buf[i]);
            if ((u >> 10) == pre22) atomicAdd(&s.Cc[u & (NB3 - 1)], 1u);
        }
    } else {
        forEach([&](float x) {
            const u32 u = fkey(x);
            if ((u >> 10) == pre22) atomicAdd(&s.Cc[u & (NB3 - 1)], 1u);
        });
    }
    __syncthreads();
    suffixScanT(s.Cc, NB3, tid, s.auxA, s.auxB);
    findBucketT(s.Cc, NB3, kT3, tid, &s.foundB, &s.foundNext);
    const u32 tkKey = (pre22 << 10) | s.foundB;
    __syncthreads();

    // ---------- P4: complete weighted L1 bucket b1 from candidates (no global read) ----------
    if (candN1 <= CANDCAP) {
        u64 lq = 0ULL;
        for (int i = tid; i < candN1; i += BDIM) {
            const float x = s.buf[i];
            if (fkey(x) >= tkKey) lq += (u64)(__expf(x - M) * SCALE);
        }
        if (lq) atomicAdd(&s.A[b1], lq);
        __syncthreads();
    } else {
        zeroA(NB1);
        __syncthreads();
        forEach([&](float x) {
            const u32 u = fkey(x);
            if (u >= tkKey) atomicAdd(&s.A[u >> 21], (u64)(__expf(x - M) * SCALE));
        });
        __syncthreads();
    }
    suffixScanT(s.A, NB1, tid, s.auxA, s.auxB);
    if (tid == 0) {
        const u64 Zu = s.A[0];
        double g = (double)prow[row] * (double)Zu;
        u64 T = (u64)ceil(g);
        if (T < 1ULL) T = 1ULL;
        if (T > Zu) T = Zu;
        s.target = T;
    }
    __syncthreads();
    const u64 wT = s.target;
    findBucketT(s.A, NB1, wT, tid, &s.foundB, &s.foundNext);
    const u32 c1  = s.foundB;
    const u64 wT2 = wT - s.foundNext;
    __syncthreads();

    // Source for weighted L2/L3: 1 = top region, 2 = bucket-b1 cands, 0 = global fallback
    const int src = (c1 >= bTop && topN <= TOPCAP) ? 1
                  : ((c1 == b1 && candN1 <= CANDCAP) ? 2 : 0);

    // ---------- P5: weighted L2 into A (reused) ----------
    zeroA(NB1);
    if (src == 0 && tid == 0) s.candCnt = 0u;   // fresh compaction for fallback
    __syncthreads();
    if (src == 1) {
        for (int j = BUFN - topN + tid; j < BUFN; j += BDIM) {
            const float x = s.buf[j];
            const u32 u = fkey(x);
            if (u >= tkKey && (u >> 21) == c1)
                atomicAdd(&s.A[(u >> 10) & (NB1 - 1)], (u64)(__expf(x - M) * SCALE));
        }
    } else if (src == 2) {
        for (int i = tid; i < candN1; i += BDIM) {
            const float x = s.buf[i];
            const u32 u = fkey(x);
            if (u >= tkKey && (u >> 21) == c1)
                atomicAdd(&s.A[(u >> 10) & (NB1 - 1)], (u64)(__expf(x - M) * SCALE));
        }
    } else {
        forEach([&](float x) {
            const u32 u = fkey(x);
            if (u >= tkKey && (u >> 21) == c1) {
                atomicAdd(&s.A[(u >> 10) & (NB1 - 1)], (u64)(__expf(x - M) * SCALE));
                const u32 idx = atomicAdd(&s.candCnt, 1u);
                if (idx < CANDCAP) s.buf[idx] = x;
            }
        });
    }
    __syncthreads();
    const int srcN = (src == 0) ? (int)s.candCnt : 0;
    suffixScanT(s.A, NB1, tid, s.auxA, s.auxB);
    findBucketT(s.A, NB1, wT2, tid, &s.foundB, &s.foundNext);
    const u32 c2  = s.foundB;
    const u64 wT3 = wT2 - s.foundNext;
    const u32 pre22w = (c1 << 11) | c2;
    __syncthreads();

    // ---------- P6: weighted L3 -> exact T_p ----------
    zeroA(NB3);
    __syncthreads();
    if (src == 1) {
        for (int j = BUFN - topN + tid; j < BUFN; j += BDIM) {
            const float x = s.buf[j];
            const u32 u = fkey(x);
            if (u >= tkKey && (u >> 10) == pre22w)
                atomicAdd(&s.A[u & (NB3 - 1)], (u64)(__expf(x - M) * SCALE));
        }
    } else if (src == 2) {
        for (int i = tid; i < candN1; i += BDIM) {
            const float x = s.buf[i];
            const u32 u = fkey(x);
            if (u >= tkKey && (u >> 10) == pre22w)
                atomicAdd(&s.A[u & (NB3 - 1)], (u64)(__expf(x - M) * SCALE));
        }
    } else if (srcN <= CANDCAP) {
        for (int i = tid; i < srcN; i += BDIM) {
            const float x = s.buf[i];                  // already kept & bucket c1
            const u32 u = fkey(x);
            if ((u >> 10) == pre22w)
                atomicAdd(&s.A[u & (NB3 - 1)], (u64)(__expf(x - M) * SCALE));
        }
    } else {
        forEach([&](float x) {
            const u32 u = fkey(x);
            if (u >= tkKey && (u >> 10) == pre22w)
                atomicAdd(&s.A[u & (NB3 - 1)], (u64)(__expf(x - M) * SCALE));
        });
    }
    __syncthreads();
    suffixScanT(s.A, NB3, tid, s.auxA, s.auxB);
    findBucketT(s.A, NB3, wT3, tid, &s.foundB, &s.foundNext);
    const u32 tpKey = (pre22w << 10) | s.foundB;
    __syncthreads();

    // ---------- P7: final elementwise mask + non-temporal store (global read #3) ----------
    const u32 thr = (tpKey > tkKey) ? tpKey : tkKey;  // provably tpKey>=tkKey; belt & braces
    const float NEGINF = -__builtin_inff();
    if (vec4) {
        const vfloat4* __restrict__ r4 = (const vfloat4*)rp;
        vfloat4* __restrict__ o4 = (vfloat4*)op;
        for (int i = tid; i < n4; i += BDIM) {
            vfloat4 v = r4[i];
            v.x = (fkey(v.x) >= thr) ? v.x : NEGINF;
            v.y = (fkey(v.y) >= thr) ? v.y : NEGINF;
            v.z = (fkey(v.z) >= thr) ? v.z : NEGINF;
            v.w = (fkey(v.w) >= thr) ? v.w : NEGINF;
            __builtin_nontemporal_store(v, &o4[i]);
        }
    } else {
        for (int i = tid; i < V; i += BDIM) {
            const float x = rp[i];
            op[i] = (fkey(x) >= thr) ? x : NEGINF;
        }
    }
}

extern "C" void kernel_launch(void* const* d_in, const int* in_sizes, int n_in,
                              void* d_out, int out_size, void* d_ws, size_t ws_size,
                              hipStream_t stream)
{
    const float* logits = (const float*)d_in[0];
    const int*   k      = (const int*)d_in[1];
    const float* p      = (const float*)d_in[2];
    float* out = (float*)d_out;

    const int B = in_sizes[1];                 // k has one entry per row
    const int V = (B > 0) ? (in_sizes[0] / B) : 0;
    if (B <= 0 || V <= 0) return;

    topktopp_mask_kernel<<<dim3(B), dim3(BDIM), 0, stream>>>(logits, k, p, out, V);

    (void)n_in; (void)out_size; (void)d_ws; (void)ws_size;
}